// GridPyramid_74998718923129
// MI455X (gfx1250) — compile-verified
//
#include <hip/hip_runtime.h>
#include <hip/hip_bf16.h>

// ---------------------------------------------------------------------------
// GridPyramid for MI455X (gfx1250, wave32).
//
// radius_search: d^2 = ||q||^2 + ||s||^2 - 2 q.s computed as rank-4 f32 WMMAs
//   (V_WMMA_F32_16X16X4_F32):  A row m (support) = (-2sx,-2sy,-2sz, ss)
//                              B col n (query)   = ( qx,  qy,  qz,  1)
//                              C                 = qq (per-query, lane-uniform)
//   => D[m][n] = d^2(support m, query n) with zero VALU fixup.
//   Queries are the N dimension, so each lane's 8 D-VGPRs are 8 candidates
//   for ONE query; the two lane halves (support rows v and v+8) keep
//   independent sorted half-lists in LDS, merged once at the end.
//   Two support tiles (32 supports) per iteration; selection is skipped with
//   a wave-uniform branch when no lane has a candidate inside the radius
//   (true for >99% of tiles at these densities).
// ---------------------------------------------------------------------------

typedef __attribute__((ext_vector_type(2))) float v2f;
typedef __attribute__((ext_vector_type(8))) float v8f;

#define KMAX 38

// ------------------------------ voxel pooling ------------------------------

__global__ void pool_clear_kernel(float* sums, int* cnt, int cap) {
    int i = blockIdx.x * blockDim.x + threadIdx.x;
    if (i < cap) {
        sums[3 * i + 0] = 0.0f;
        sums[3 * i + 1] = 0.0f;
        sums[3 * i + 2] = 0.0f;
        cnt[i] = 0;
    }
}

__global__ void pool_insert_kernel(const float* __restrict__ pts, int n,
                                   float inv_vs, int dim,
                                   float* __restrict__ sums,
                                   int* __restrict__ cnt, int cap) {
    int i = blockIdx.x * blockDim.x + threadIdx.x;
    if (i >= n) return;
    float x = pts[3 * i + 0];
    float y = pts[3 * i + 1];
    float z = pts[3 * i + 2];
    if (x >= 1e5f) return;  // sentinel point from previous level
    int cx = min(max((int)floorf(x * inv_vs), 0), dim - 1);
    int cy = min(max((int)floorf(y * inv_vs), 0), dim - 1);
    int cz = min(max((int)floorf(z * inv_vs), 0), dim - 1);
    int slot = (cx * dim + cy) * dim + cz;  // sorted-key-order, deterministic
    slot = min(slot, cap - 1);
    atomicAdd(&sums[3 * slot + 0], x);
    atomicAdd(&sums[3 * slot + 1], y);
    atomicAdd(&sums[3 * slot + 2], z);
    atomicAdd(&cnt[slot], 1);
}

__global__ void pool_final_kernel(const float* __restrict__ sums,
                                  const int* __restrict__ cnt,
                                  float* __restrict__ outp, int cap) {
    int i = blockIdx.x * blockDim.x + threadIdx.x;
    if (i >= cap) return;
    int c = cnt[i];
    float inv = 1.0f / (float)max(c, 1);
    bool ok = c > 0;
    outp[3 * i + 0] = ok ? sums[3 * i + 0] * inv : 1e6f;
    outp[3 * i + 1] = ok ? sums[3 * i + 1] * inv : 1e6f;
    outp[3 * i + 2] = ok ? sums[3 * i + 2] * inv : 1e6f;
}

// ---------------------------- WMMA radius search ---------------------------
// Block = 128 threads = 4 waves. Each wave owns 16 queries; both lane halves
// work on the same 16 queries but on different support rows of each WMMA tile.

__device__ __forceinline__ void insert_candidates(const v8f& d, int nb_base,
                                                  float r2, int k, int& cnt,
                                                  float* DL, int* IL) {
    // wave-uniform early-out: skip selection when no lane has a candidate
    float m01 = fminf(d[0], d[1]);
    float m23 = fminf(d[2], d[3]);
    float m45 = fminf(d[4], d[5]);
    float m67 = fminf(d[6], d[7]);
    float dmin = fminf(fminf(m01, m23), fminf(m45, m67));
    if (__any(dmin <= r2)) {
#pragma unroll
        for (int v = 0; v < 8; ++v) {
            float dd = d[v];
            if (dd <= r2) {  // padded tail supports always fail this
                if (!(cnt == k && dd >= DL[k - 1])) {
                    int p = (cnt < k) ? cnt : (k - 1);  // cap: drop current max
                    while (p > 0 && DL[p - 1] > dd) {
                        DL[p] = DL[p - 1];
                        IL[p] = IL[p - 1];
                        --p;
                    }
                    DL[p] = dd;
                    IL[p] = nb_base + v;
                    if (cnt < k) ++cnt;
                }
            }
        }
    }
}

__global__ __launch_bounds__(128) void rsearch_wmma_kernel(
    const float* __restrict__ q, int nq, const float* __restrict__ s, int ns,
    float r2, int k, int* __restrict__ out) {
    __shared__ float sdist[4][32][KMAX];
    __shared__ int sidxs[4][32][KMAX];
    __shared__ int scnt[4][32];

    const int lane = threadIdx.x & 31;
    const int wave = threadIdx.x >> 5;
    const int n16 = lane & 15;   // query column within tile
    const int hi = lane >> 4;    // lane-half: selects K{2,3} and rows +8
    const int qbase = (blockIdx.x * 4 + wave) * 16;
    const int qi = qbase + n16;

    // --- query fragment (B matrix, 4x16) + C = qq, built once ---
    int qidx = min(qi, nq - 1);
    bool qvalid = qi < nq;
    float qx = q[3 * qidx + 0];
    float qy = q[3 * qidx + 1];
    float qz = q[3 * qidx + 2];
    qx = qvalid ? qx : 3.0e7f;  // out-of-range query: never matches anything
    qy = qvalid ? qy : 3.0e7f;
    qz = qvalid ? qz : 3.0e7f;
    float qq = qx * qx + qy * qy + qz * qz;

    v2f bfrag;
    bfrag.x = hi ? qz : qx;    // K = 2 : 0
    bfrag.y = hi ? 1.0f : qy;  // K = 3 : 1
    v8f cfrag = {qq, qq, qq, qq, qq, qq, qq, qq};

    float* DL = sdist[wave][lane];
    int* IL = sidxs[wave][lane];
    int cnt = 0;  // entries in this lane's half-list

    for (int sbase = 0; sbase < ns; sbase += 32) {
        // two 16-support tiles per iteration
        int si0 = sbase + n16;
        int si1 = sbase + 16 + n16;
        int sx0i = min(si0, ns - 1);
        int sx1i = min(si1, ns - 1);
        __builtin_prefetch(&s[(size_t)min(si0 + 128, ns - 1) * 3], 0, 1);

        float ax = s[3 * sx0i + 0], ay = s[3 * sx0i + 1], az = s[3 * sx0i + 2];
        float bx = s[3 * sx1i + 0], by = s[3 * sx1i + 1], bz = s[3 * sx1i + 2];
        bool v0 = si0 < ns, v1 = si1 < ns;
        ax = v0 ? ax : 9.0e6f; ay = v0 ? ay : 9.0e6f; az = v0 ? az : 9.0e6f;
        bx = v1 ? bx : 9.0e6f; by = v1 ? by : 9.0e6f; bz = v1 ? bz : 9.0e6f;
        float ssa = ax * ax + ay * ay + az * az;
        float ssb = bx * bx + by * by + bz * bz;

        // A fragments (supports, 16x4): row = (-2sx, -2sy, -2sz, ss)
        v2f af0, af1;
        af0.x = hi ? (-2.0f * az) : (-2.0f * ax);  // K = 2 : 0
        af0.y = hi ? ssa : (-2.0f * ay);           // K = 3 : 1
        af1.x = hi ? (-2.0f * bz) : (-2.0f * bx);
        af1.y = hi ? ssb : (-2.0f * by);

        // D[m][n] = ss - 2 s.q + qq = d^2   (EXEC all-ones here; independent)
        v8f d0 = __builtin_amdgcn_wmma_f32_16x16x4_f32(
            false, af0, false, bfrag, (short)0, cfrag, false, false);
        v8f d1 = __builtin_amdgcn_wmma_f32_16x16x4_f32(
            false, af1, false, bfrag, (short)0, cfrag, false, false);

        // lane holds d^2 for supports (base + v), v=0..7, of its query n16
        insert_candidates(d0, sbase + 8 * hi, r2, k, cnt, DL, IL);
        insert_candidates(d1, sbase + 16 + 8 * hi, r2, k, cnt, DL, IL);
    }

    scnt[wave][lane] = cnt;
    __syncthreads();  // make partner half-lists visible

    // lanes 0..15 merge the two sorted half-lists of their query, pad with ns
    if (lane < 16 && qi < nq) {
        const int c0 = scnt[wave][lane];
        const int c1 = scnt[wave][lane + 16];
        const float* D0 = sdist[wave][lane];
        const int* I0 = sidxs[wave][lane];
        const float* D1 = sdist[wave][lane + 16];
        const int* I1 = sidxs[wave][lane + 16];
        int* orow = out + (size_t)qi * k;
        int i0 = 0, i1 = 0;
        for (int j = 0; j < k; ++j) {
            int val = ns;
            if (i0 < c0 && (i1 >= c1 || D0[i0] <= D1[i1]))
                val = I0[i0++];
            else if (i1 < c1)
                val = I1[i1++];
            orow[j] = val;
        }
    }
}

// ------------------------------ host launcher ------------------------------

extern "C" void kernel_launch(void* const* d_in, const int* in_sizes, int n_in,
                              void* d_out, int out_size, void* d_ws,
                              size_t ws_size, hipStream_t stream) {
    (void)in_sizes; (void)n_in; (void)out_size; (void)ws_size;

    constexpr int N0 = 16384, N1 = 8192, N2 = 4096;
    constexpr int K0 = 26, K1 = 31, K2 = 38;
    // flat element offsets into d_out, in reference return order
    constexpr size_t O_PTS0 = 0;
    constexpr size_t O_PTS1 = O_PTS0 + (size_t)N0 * 3;
    constexpr size_t O_PTS2 = O_PTS1 + (size_t)N1 * 3;
    constexpr size_t O_NB0 = O_PTS2 + (size_t)N2 * 3;
    constexpr size_t O_NB1 = O_NB0 + (size_t)N0 * K0;
    constexpr size_t O_NB2 = O_NB1 + (size_t)N1 * K1;
    constexpr size_t O_SUB0 = O_NB2 + (size_t)N2 * K2;
    constexpr size_t O_SUB1 = O_SUB0 + (size_t)N1 * K0;
    constexpr size_t O_UP0 = O_SUB1 + (size_t)N2 * K1;
    constexpr size_t O_UP1 = O_UP0 + (size_t)N0 * 1;

    const float* pts0 = (const float*)d_in[0];
    float* out_f = (float*)d_out;
    int* out_i = (int*)d_out;

    // workspace: two direct-mapped voxel accumulators
    float* sums1 = (float*)d_ws;                    // N1*3 floats
    int* cnt1 = (int*)(sums1 + (size_t)N1 * 3);     // N1 ints
    float* sums2 = (float*)(cnt1 + N1);             // N2*3 floats
    int* cnt2 = (int*)(sums2 + (size_t)N2 * 3);     // N2 ints

    // pts0 passthrough
    hipMemcpyAsync(out_f + O_PTS0, pts0, (size_t)N0 * 3 * sizeof(float),
                   hipMemcpyDeviceToDevice, stream);

    // level 1 pooling: voxel 0.2, grid 20^3 = 8000 <= 8192 slots
    pool_clear_kernel<<<(N1 + 255) / 256, 256, 0, stream>>>(sums1, cnt1, N1);
    pool_insert_kernel<<<(N0 + 255) / 256, 256, 0, stream>>>(
        pts0, N0, 1.0f / 0.2f, 20, sums1, cnt1, N1);
    pool_final_kernel<<<(N1 + 255) / 256, 256, 0, stream>>>(
        sums1, cnt1, out_f + O_PTS1, N1);

    // level 2 pooling: voxel 0.4, grid 10^3 = 1000 <= 4096 slots
    pool_clear_kernel<<<(N2 + 255) / 256, 256, 0, stream>>>(sums2, cnt2, N2);
    pool_insert_kernel<<<(N1 + 255) / 256, 256, 0, stream>>>(
        out_f + O_PTS1, N1, 1.0f / 0.4f, 10, sums2, cnt2, N2);
    pool_final_kernel<<<(N2 + 255) / 256, 256, 0, stream>>>(
        sums2, cnt2, out_f + O_PTS2, N2);

    const float* p1 = out_f + O_PTS1;
    const float* p2 = out_f + O_PTS2;

    auto rs = [&](const float* q, int nq, const float* s, int ns, float r,
                  int k, int* o) {
        dim3 grid((nq + 63) / 64), block(128);
        rsearch_wmma_kernel<<<grid, block, 0, stream>>>(q, nq, s, ns, r * r, k, o);
    };

    // intra-level neighbors
    rs(pts0, N0, pts0, N0, 0.25f, K0, out_i + O_NB0);
    rs(p1, N1, p1, N1, 0.50f, K1, out_i + O_NB1);
    rs(p2, N2, p2, N2, 1.00f, K2, out_i + O_NB2);
    // subsampling: coarse queries over fine supports, fine radius
    rs(p1, N1, pts0, N0, 0.25f, K0, out_i + O_SUB0);
    rs(p2, N2, p1, N1, 0.50f, K1, out_i + O_SUB1);
    // upsampling: nearest coarse point within coarse radius
    rs(pts0, N0, p1, N1, 0.50f, 1, out_i + O_UP0);
    rs(p1, N1, p2, N2, 1.00f, 1, out_i + O_UP1);
}